// SpatialDualDescriptorRN3_85882166050871
// MI455X (gfx1250) — compile-verified
//
#include <hip/hip_runtime.h>
#include <stdint.h>

typedef float v2f __attribute__((ext_vector_type(2)));
typedef float v4f __attribute__((ext_vector_type(4)));
typedef float v8f __attribute__((ext_vector_type(8)));

#define MDIM 64
#define LDIM 512
#define NBLOCKS 512
#define TPB 256

// ---------------------------------------------------------------------------
// Kernel 1: transpose Acoeff[64][512] -> At[512][64] so per-position column
// gathers become contiguous 256B reads. 32768 elements, one-shot, negligible.
// ---------------------------------------------------------------------------
__global__ void transpose_acoeff_kernel(const float* __restrict__ A,
                                        float* __restrict__ At) {
    int i = blockIdx.x * blockDim.x + threadIdx.x;   // 0..32767
    if (i < LDIM * MDIM) {
        int j = i >> 6;        // column of Acoeff (0..511)
        int m = i & 63;        // row of Acoeff   (0..63)
        At[i] = A[m * LDIM + j];
    }
}

// ---------------------------------------------------------------------------
// Kernel 2: S[r][v] = sum_m Bcat[r][m] * E[v][m]   (24x26, K=64)
// via V_WMMA_F32_16X16X4_F32: 2x2 tiles of 16x16, 16 K-steps, one wave.
// Bcat rows: 0-7 = B_x, 8-15 = B_y, 16-23 = B_z; rows>=24, cols>=26 zero-pad.
// Branchless boundary handling: clamped index + 0/1 mask (no exec divergence).
// S stored padded as [24][32] for cheap indexing in the main kernel.
// ---------------------------------------------------------------------------
__device__ __forceinline__ float bcat_val(int r, int k,
                                          const float* __restrict__ Bx,
                                          const float* __restrict__ By,
                                          const float* __restrict__ Bz) {
    float mask = (r < 24) ? 1.0f : 0.0f;
    int rc = (r < 24) ? r : 23;
    const float* base = (rc < 8) ? Bx : ((rc < 16) ? By : Bz);
    int rr = (rc < 8) ? rc : ((rc < 16) ? rc - 8 : rc - 16);
    return base[rr * MDIM + k] * mask;
}

__global__ void stab_wmma_kernel(const float* __restrict__ E,
                                 const float* __restrict__ Bx,
                                 const float* __restrict__ By,
                                 const float* __restrict__ Bz,
                                 float* __restrict__ Sg) {
    const int lane = threadIdx.x;        // 32 threads, 1 wave
    const int half = lane >> 4;          // 0 or 1
    const int l16  = lane & 15;

    v8f acc[2][2] = {};

    for (int kb = 0; kb < MDIM; kb += 4) {
        const int kk = kb + half * 2;    // K index of VGPR0 for this lane

        // A fragments (16x4 f32): lane l, vgpr i -> (M = l%16, K = (l/16)*2+i)
        v2f afrag[2];
        #pragma unroll
        for (int ti = 0; ti < 2; ++ti) {
            int r = ti * 16 + l16;
            v2f a;
            a.x = bcat_val(r, kk, Bx, By, Bz);
            a.y = bcat_val(r, kk + 1, Bx, By, Bz);
            afrag[ti] = a;
        }
        // B fragments (4x16 f32): lane l, vgpr i -> (K = (l/16)*2+i, N = l%16)
        // B[k][n] = E[n][k]  (we need E^T as the right operand)
        v2f bfrag[2];
        #pragma unroll
        for (int tj = 0; tj < 2; ++tj) {
            int v = tj * 16 + l16;
            float mask = (v < 26) ? 1.0f : 0.0f;
            int vc = (v < 26) ? v : 25;
            v2f b;
            b.x = E[vc * MDIM + kk]     * mask;
            b.y = E[vc * MDIM + kk + 1] * mask;
            bfrag[tj] = b;
        }
        #pragma unroll
        for (int ti = 0; ti < 2; ++ti)
            #pragma unroll
            for (int tj = 0; tj < 2; ++tj)
                acc[ti][tj] = __builtin_amdgcn_wmma_f32_16x16x4_f32(
                    /*neg_a=*/false, afrag[ti],
                    /*neg_b=*/false, bfrag[tj],
                    /*c_mod=*/(short)0, acc[ti][tj],
                    /*reuse_a=*/false, /*reuse_b=*/false);
    }

    // D layout: lane l, vgpr i -> (M = 8*(l/16)+i, N = l%16)
    #pragma unroll
    for (int ti = 0; ti < 2; ++ti) {
        #pragma unroll
        for (int tj = 0; tj < 2; ++tj) {
            #pragma unroll
            for (int i = 0; i < 8; ++i) {
                int r = ti * 16 + half * 8 + i;
                int v = tj * 16 + l16;
                if (r < 24 && v < 32)
                    Sg[r * 32 + v] = acc[ti][tj][i];   // padded cols hold 0
            }
        }
    }
}

// ---------------------------------------------------------------------------
// Kernel 3 (store-bound main loop).
// LDS: At (128KB) + S (3KB) + per-batch (scalar,j) (2KB) ~= 133KB -> 2 blk/WGP.
// At staged with GLOBAL_LOAD_ASYNC_TO_LDS_B128 (memory->LDS DMA, ASYNCcnt).
// Phase 1: 256 threads compute (scalar, j) for 256 positions (coalesced loads).
// Phase 2: 16 threads/row scale the LDS-resident A-column; each wave emits
//          512B of contiguous non-temporal stores (out stream never re-read).
// ---------------------------------------------------------------------------
__global__ void __launch_bounds__(TPB)
spatial_dual_main_kernel(const int* __restrict__ tok,
                         const int* __restrict__ k1,
                         const int* __restrict__ k2,
                         const int* __restrict__ k3,
                         const float* __restrict__ Atg,
                         const float* __restrict__ Sg,
                         float* __restrict__ out, int N) {
    __shared__ float lAt[LDIM * MDIM];   // 131072 B
    __shared__ float lS[24 * 32];        //   3072 B
    __shared__ float sc[TPB];
    __shared__ int   sj[TPB];

    const int t = threadIdx.x;

    // Stage At into LDS via async global->LDS 128-bit copies (no VGPR bounce).
    // LDS byte offset = low 32 bits of the flat shared-aperture address.
    {
        const uint64_t gbase = (uint64_t)(uintptr_t)Atg;
        #pragma unroll 4
        for (int i = t; i < LDIM * MDIM / 4; i += TPB) {
            unsigned lds_off = (unsigned)(uintptr_t)(&lAt[i * 4]);
            uint64_t gaddr   = gbase + (uint64_t)i * 16u;
            asm volatile("global_load_async_to_lds_b128 %0, %1, off"
                         :: "v"(lds_off), "v"(gaddr)
                         : "memory");
        }
        for (int i = t; i < 24 * 32; i += TPB) lS[i] = Sg[i];
        asm volatile("s_wait_asynccnt 0x0" ::: "memory");
    }
    __syncthreads();

    const int batches = (N + TPB - 1) / TPB;
    const int g  = t & 15;     // column group: floats 4g..4g+3
    const int pr = t >> 4;     // row-within-iteration

    for (int b = blockIdx.x; b < batches; b += gridDim.x) {
        const int n0 = b * TPB;
        // phase 1: per-position scalar + flattened basis index
        {
            int n = n0 + t;
            if (n < N) {
                int tk = tok[n];
                int j1 = k1[n] & 7, j2 = k2[n] & 7, j3 = k3[n] & 7;
                float s = lS[j1 * 32 + tk]
                        * lS[(8 + j2) * 32 + tk]
                        * lS[(16 + j3) * 32 + tk];
                sc[t] = s;
                sj[t] = (j1 << 6) | (j2 << 3) | j3;
            } else {
                sc[t] = 0.0f;
                sj[t] = 0;
            }
        }
        __syncthreads();
        // phase 2: stream 256 rows x 256B, fully coalesced NT stores
        #pragma unroll 4
        for (int it = 0; it < 16; ++it) {
            int p = it * 16 + pr;            // 0..255
            int n = n0 + p;
            if (n < N) {
                float s = sc[p];             // LDS broadcast
                int   j = sj[p];
                const v4f a = *(const v4f*)&lAt[j * MDIM + g * 4];
                v4f o = a * s;
                v4f* dptr = (v4f*)(out + (size_t)n * MDIM + g * 4);
                __builtin_nontemporal_store(o, dptr);
            }
        }
        __syncthreads();
    }
}

extern "C" void kernel_launch(void* const* d_in, const int* in_sizes, int n_in,
                              void* d_out, int out_size, void* d_ws, size_t ws_size,
                              hipStream_t stream) {
    const float* E   = (const float*)d_in[0];   // embedding [26,64]
    const float* Ac  = (const float*)d_in[1];   // Acoeff   [64,512]
    const float* Bx  = (const float*)d_in[2];   // [8,64]
    const float* By  = (const float*)d_in[3];   // [8,64]
    const float* Bz  = (const float*)d_in[4];   // [8,64]
    const int*   tok = (const int*)d_in[5];     // [N]
    const int*   k1  = (const int*)d_in[6];
    const int*   k2  = (const int*)d_in[7];
    const int*   k3  = (const int*)d_in[8];
    float* out = (float*)d_out;
    const int N = in_sizes[5];

    float* At = (float*)d_ws;                   // [512][64] = 128 KB
    float* S  = At + LDIM * MDIM;               // [24][32]  =   3 KB

    transpose_acoeff_kernel<<<(LDIM * MDIM + TPB - 1) / TPB, TPB, 0, stream>>>(Ac, At);
    stab_wmma_kernel<<<1, 32, 0, stream>>>(E, Bx, By, Bz, S);
    spatial_dual_main_kernel<<<NBLOCKS, TPB, 0, stream>>>(tok, k1, k2, k3, At, S, out, N);
}